// RaeExt_69114613730753
// MI455X (gfx1250) — compile-verified
//
#include <hip/hip_runtime.h>

typedef float v2f __attribute__((ext_vector_type(2)));
typedef float v8f __attribute__((ext_vector_type(8)));

#define BQ    2048   // batch of queries
#define TOPK  64
#define LDIM  256
#define LDA   260    // padded LDS stride for A tile (260 % 64 == 4 -> no bank conflicts)

// ---------------------------------------------------------------------------
// Kernel A: per-query softmax over TOPK sims + weighted gather-reduction of
// raw value rows:  g[b][l] = sum_k softmax(sim[b]*10)[k] * V[idx[b][k]][l]
// One 256-thread block per query b; thread == output column l.
// (out = (sum_k w_k * V[idx_k]) @ T  ==  reference's einsum over (V@T)[idx],
//  so we avoid the 13.1 GFLOP dense 100000x256x256 GEMM entirely.)
// ---------------------------------------------------------------------------
__global__ __launch_bounds__(256) void rae_softmax_gather_kernel(
    const float* __restrict__ sim,      // [BQ, TOPK]
    const float* __restrict__ values,   // [K_DIM, LDIM]
    const int*   __restrict__ indices,  // [BQ, TOPK]
    float*       __restrict__ g)        // [BQ, LDIM] (workspace)
{
  __shared__ float w[TOPK];
  __shared__ float red[TOPK];
  __shared__ int   idx[TOPK];

  const int b   = blockIdx.x;
  const int tid = threadIdx.x;

  if (tid < TOPK) {
    float v = sim[b * TOPK + tid] * 10.0f;   // 1/TEMPERATURE = 10
    w[tid]   = v;
    red[tid] = v;
    idx[tid] = indices[b * TOPK + tid];
  }
  __syncthreads();

  // max-reduce over 64 entries
  #pragma unroll
  for (int off = 32; off >= 1; off >>= 1) {
    if (tid < off) red[tid] = fmaxf(red[tid], red[tid + off]);
    __syncthreads();
  }
  const float mx = red[0];
  __syncthreads();

  if (tid < TOPK) {
    float e = expf(w[tid] - mx);
    w[tid]   = e;
    red[tid] = e;
  }
  __syncthreads();

  // sum-reduce
  #pragma unroll
  for (int off = 32; off >= 1; off >>= 1) {
    if (tid < off) red[tid] += red[tid + off];
    __syncthreads();
  }
  const float inv = 1.0f / red[0];
  __syncthreads();

  if (tid < TOPK) w[tid] *= inv;
  __syncthreads();

  // Weighted gather-sum. For each k all 256 threads read one contiguous 1KB
  // row of `values` -> fully coalesced; values (100MB) is L2-resident (192MB).
  float acc = 0.0f;
  #pragma unroll 8
  for (int k = 0; k < TOPK; ++k) {
    acc = fmaf(w[k], values[(size_t)idx[k] * LDIM + tid], acc);
  }
  g[(size_t)b * LDIM + tid] = acc;
}

// ---------------------------------------------------------------------------
// Kernel B: out[2048,256] = g[2048,256] @ T[256,256] in fp32 using
// V_WMMA_F32_16X16X4_F32.  Block = 128 threads (4 waves). Each wave computes
// one 16x16 output tile; the block covers a 16(M) x 64(N) region.
//
// Fragment layouts (CDNA5 ISA 7.12.2, 32-bit):
//   A 16x4 : lanes 0-15 -> row M=lane, {VGPR0,VGPR1} = K{0,1};
//            lanes 16-31 -> row M=lane-16, {VGPR0,VGPR1} = K{2,3}
//   B 4x16 : mirrored (lanes 0-15: rows K{0,1}, col N=lane;
//            lanes 16-31: rows K{2,3}, col N=lane-16)
//   C/D    : VGPR i -> M=i (lanes 0-15) / M=8+i (lanes 16-31), N = lane%16
// ---------------------------------------------------------------------------
__global__ __launch_bounds__(128) void rae_gemm_wmma_kernel(
    const float* __restrict__ g,    // [BQ, LDIM]
    const float* __restrict__ T,    // [LDIM, LDIM]
    float*       __restrict__ out)  // [BQ, LDIM]
{
  __shared__ float ldsA[16 * LDA];       // 16 rows of g, padded stride
  __shared__ float ldsB[LDIM * 64];      // 256 x 64 slice of T

  const int tid = threadIdx.x;
  const int m0  = blockIdx.x * 16;
  const int n0  = blockIdx.y * 64;

  // Stage A tile (16 x 256) with float4 loads; rows m0..m0+15 are contiguous.
  // Unroll capped: keeps <= 4 float4 in flight per thread so total VGPR
  // allocation stays well inside the 256 directly-addressable window.
  {
    const float4* src = (const float4*)(g + (size_t)m0 * LDIM);
    #pragma unroll 4
    for (int i = tid; i < 16 * 64; i += 128) {
      int row = i >> 6;        // 64 float4 per row
      int c4  = i & 63;
      *(float4*)&ldsA[row * LDA + c4 * 4] = src[i];
    }
  }
  // Stage B tile (256 x 64): rows of T, columns n0..n0+63.
  #pragma unroll 4
  for (int i = tid; i < LDIM * 16; i += 128) {
    int row = i >> 4;          // 16 float4 per row
    int c4  = i & 15;
    *(float4*)&ldsB[row * 64 + c4 * 4] =
        *(const float4*)(T + (size_t)row * LDIM + n0 + c4 * 4);
  }
  __syncthreads();

  const int wave = tid >> 5;
  const int lane = tid & 31;
  const int hi   = lane >> 4;          // 0 or 1 (K sub-pair selector)
  const int lo   = lane & 15;          // M row / N col within tile
  const int col  = wave * 16 + lo;     // column within the 64-wide B tile

  v8f c = {};
  #pragma unroll 4
  for (int k = 0; k < LDIM / 4; ++k) {
    const int kk = 4 * k + 2 * hi;
    v2f a = *(const v2f*)&ldsA[lo * LDA + kk];   // 8B aligned (LDA, kk even)
    v2f bv;
    bv.x = ldsB[kk * 64 + col];
    bv.y = ldsB[(kk + 1) * 64 + col];
    // D = A(16x4) * B(4x16) + C ; 8 args: neg_a, A, neg_b, B, c_mod, C, reuse_a, reuse_b
    c = __builtin_amdgcn_wmma_f32_16x16x4_f32(false, a, false, bv,
                                              (short)0, c, false, false);
  }

  #pragma unroll
  for (int i = 0; i < 8; ++i) {
    out[(size_t)(m0 + hi * 8 + i) * LDIM + n0 + col] = c[i];
  }
}

// ---------------------------------------------------------------------------
// Inputs (setup_inputs order): sim f32[2048*64], values f32[100000*256],
// trainable f32[256*256], indices i32[2048*64]. Output f32[2048*256].
// ---------------------------------------------------------------------------
extern "C" void kernel_launch(void* const* d_in, const int* in_sizes, int n_in,
                              void* d_out, int out_size, void* d_ws, size_t ws_size,
                              hipStream_t stream) {
  const float* sim     = (const float*)d_in[0];
  const float* values  = (const float*)d_in[1];
  const float* trainT  = (const float*)d_in[2];
  const int*   indices = (const int*)d_in[3];
  float*       out     = (float*)d_out;
  float*       g       = (float*)d_ws;   // 2048*256*4 = 2 MB scratch

  rae_softmax_gather_kernel<<<BQ, 256, 0, stream>>>(sim, values, indices, g);

  dim3 grid(BQ / 16, LDIM / 64);
  rae_gemm_wmma_kernel<<<grid, 128, 0, stream>>>(g, trainT, out);
}